// GCNEncoder_46093589021376
// MI455X (gfx1250) — compile-verified
//
#include <hip/hip_runtime.h>

typedef __attribute__((ext_vector_type(2))) float v2f;
typedef __attribute__((ext_vector_type(8))) float v8f;

#define IN_F  128
#define HID_F 128
#define OUT_F 64

// ---------------- utility: fill ----------------
__global__ void k_fill(float* __restrict__ p, float v, long n) {
    long i = (long)blockIdx.x * blockDim.x + threadIdx.x;
    long stride = (long)gridDim.x * blockDim.x;
    for (; i < n; i += stride) p[i] = v;
}

// ---------------- degree (deg pre-initialized to 1.0 for self-loop) ----------------
__global__ void k_degree(const int* __restrict__ dst, float* __restrict__ deg, long E) {
    long i = (long)blockIdx.x * blockDim.x + threadIdx.x;
    long stride = (long)gridDim.x * blockDim.x;
    for (; i < E; i += stride)
        __hip_atomic_fetch_add(&deg[dst[i]], 1.0f, __ATOMIC_RELAXED, __HIP_MEMORY_SCOPE_AGENT);
}

// ---------------- dinv = rsqrt(deg), in place (deg >= 1 always) ----------------
__global__ void k_rsqrt(float* __restrict__ deg, int N) {
    int i = blockIdx.x * blockDim.x + threadIdx.x;
    if (i < N) deg[i] = rsqrtf(deg[i]);
}

// ---------------- WMMA f32 GEMM: H[M x NCOL] = X[M x K] @ W[K x NCOL] ----------------
// Each block covers RT*16 rows; NCOL/16 waves, each wave owns one 16-col strip and
// keeps RT independent accumulators, so every B (weight) fragment is reused RT times
// and the RT WMMAs per K-step are hazard-independent (ISA 7.12.1).
template <int K, int NCOL, int RT>
__global__ void __launch_bounds__(32 * (NCOL / 16))
k_gemm_wmma(const float* __restrict__ X, const float* __restrict__ W,
            float* __restrict__ H, int M) {
    constexpr int ROWS = 16 * RT;
    constexpr int SXS  = K + 4;          // padded LDS stride: A-lanes hit distinct banks
    __shared__ float sX[ROWS * SXS];

    const int tid  = threadIdx.x;
    const int wave = tid >> 5;
    const int lane = tid & 31;
    const int row0 = blockIdx.x * ROWS;

    // cooperative load of the ROWS x K X-tile into LDS (coalesced)
    for (int i = tid; i < ROWS * K; i += blockDim.x) {
        int r = i / K, c = i % K;
        int gr = row0 + r;
        sX[r * SXS + c] = (gr < M) ? X[(size_t)gr * K + c] : 0.0f;
    }
    __syncthreads();

    const int half = lane >> 4;          // 0: K pair {0,1}; 1: K pair {2,3}
    const int idx  = lane & 15;          // M index for A, N index for B/D
    const int colbase = wave * 16;

    v8f acc[RT] = {};
#pragma unroll
    for (int k0 = 0; k0 < K; k0 += 4) {
        const int ka = k0 + half * 2;
        v2f b;
        b.x = W[(size_t)(ka)     * NCOL + colbase + idx];
        b.y = W[(size_t)(ka + 1) * NCOL + colbase + idx];
#pragma unroll
        for (int t = 0; t < RT; ++t) {
            v2f a;
            a.x = sX[(t * 16 + idx) * SXS + ka];
            a.y = sX[(t * 16 + idx) * SXS + ka + 1];
            // D = A x B + C, full f32
            acc[t] = __builtin_amdgcn_wmma_f32_16x16x4_f32(
                /*neg_a=*/false, a, /*neg_b=*/false, b,
                /*c_mod=*/(short)0, acc[t], /*reuse_a=*/false, /*reuse_b=*/false);
        }
    }

    // D layout: VGPR v -> row = v + 8*half, col = idx (within tile)
#pragma unroll
    for (int t = 0; t < RT; ++t) {
#pragma unroll
        for (int v = 0; v < 8; ++v) {
            int gr = row0 + t * 16 + half * 8 + v;
            if (gr < M) H[(size_t)gr * NCOL + colbase + idx] = acc[t][v];
        }
    }
}

// ---------------- edge scatter: agg[dst] += h[src] * dinv[src]*dinv[dst] ----------------
// One wave per edge; each lane handles F/32 consecutive features (4 for F=128 -> b128 load).
template <int F>
__global__ void k_scatter(const int* __restrict__ src, const int* __restrict__ dst,
                          const float* __restrict__ dinv, const float* __restrict__ h,
                          float* __restrict__ agg, long E) {
    constexpr int C = F / 32;
    const int wavesPerBlock = blockDim.x >> 5;
    const long e = (long)blockIdx.x * wavesPerBlock + (threadIdx.x >> 5);
    if (e >= E) return;
    const int lane = threadIdx.x & 31;
    const int s = src[e];
    const int d = dst[e];
    const float nm = dinv[s] * dinv[d];
    const float* __restrict__ hp = h + (size_t)s * F + lane * C;
    float* __restrict__ op = agg + (size_t)d * F + lane * C;
    float v[C];
#pragma unroll
    for (int j = 0; j < C; ++j) v[j] = hp[j];      // contiguous -> global_load_b128/b64
#pragma unroll
    for (int j = 0; j < C; ++j)
        __hip_atomic_fetch_add(op + j, v[j] * nm, __ATOMIC_RELAXED, __HIP_MEMORY_SCOPE_AGENT);
}

// ---------------- layer-1 epilogue: self-loop + bias + ReLU (in place on agg1) ----------------
__global__ void k_post_relu(float* __restrict__ agg, const float* __restrict__ h,
                            const float* __restrict__ dinv, const float* __restrict__ b,
                            long total) {
    long i = (long)blockIdx.x * blockDim.x + threadIdx.x;
    if (i >= total) return;
    int n = (int)(i / HID_F), f = (int)(i % HID_F);
    float di = dinv[n];
    float v = agg[i] + h[i] * di * di + b[f];
    agg[i] = v > 0.0f ? v : 0.0f;
}

// ---------------- layer-2 epilogue: self-loop + bias (in place on out) ----------------
__global__ void k_post(float* __restrict__ out, const float* __restrict__ h2,
                       const float* __restrict__ dinv, const float* __restrict__ b,
                       long total) {
    long i = (long)blockIdx.x * blockDim.x + threadIdx.x;
    if (i >= total) return;
    int n = (int)(i / OUT_F), f = (int)(i % OUT_F);
    float di = dinv[n];
    out[i] = out[i] + h2[i] * di * di + b[f];
}

extern "C" void kernel_launch(void* const* d_in, const int* in_sizes, int n_in,
                              void* d_out, int out_size, void* d_ws, size_t ws_size,
                              hipStream_t stream) {
    const float* x  = (const float*)d_in[0];
    const int*   ei = (const int*)d_in[1];   // edge_index flat (2, E) row-major
    const float* W1 = (const float*)d_in[2];
    const float* b1 = (const float*)d_in[3];
    const float* W2 = (const float*)d_in[4];
    const float* b2 = (const float*)d_in[5];
    float* out = (float*)d_out;

    const int  N = in_sizes[0] / IN_F;
    const long E = (long)in_sizes[1] / 2;
    const int* srcIdx = ei;
    const int* dstIdx = ei + E;

    // workspace carve-out (floats): dinv[N] | h[N*128] | agg1[N*128] | h2[N*64]
    float* ws   = (float*)d_ws;
    float* dinv = ws;
    float* h    = dinv + N;
    float* agg1 = h + (size_t)N * HID_F;
    float* h2   = agg1 + (size_t)N * HID_F;

    // init (fresh every call -> deterministic)
    k_fill<<<1024, 256, 0, stream>>>(dinv, 1.0f, (long)N);              // self-loop deg
    k_fill<<<4096, 256, 0, stream>>>(agg1, 0.0f, (long)N * HID_F);
    k_fill<<<2048, 256, 0, stream>>>(out, 0.0f, (long)N * OUT_F);

    // symmetric normalization
    k_degree<<<4096, 256, 0, stream>>>(dstIdx, dinv, E);
    k_rsqrt<<<(N + 255) / 256, 256, 0, stream>>>(dinv, N);

    constexpr int RT   = 4;        // row tiles per block
    constexpr int ROWS = 16 * RT;  // 64 rows per block

    // layer 1: h = x @ W1 ; scatter ; +self-loop +b1, ReLU
    k_gemm_wmma<IN_F, HID_F, RT>
        <<<(N + ROWS - 1) / ROWS, 32 * (HID_F / 16), 0, stream>>>(x, W1, h, N);
    k_scatter<HID_F><<<(int)((E + 7) / 8), 256, 0, stream>>>(srcIdx, dstIdx, dinv, h, agg1, E);
    k_post_relu<<<(int)(((long)N * HID_F + 255) / 256), 256, 0, stream>>>(agg1, h, dinv, b1,
                                                                          (long)N * HID_F);

    // layer 2: h2 = h1 @ W2 ; scatter into out ; +self-loop +b2
    k_gemm_wmma<HID_F, OUT_F, RT>
        <<<(N + ROWS - 1) / ROWS, 32 * (OUT_F / 16), 0, stream>>>(agg1, W2, h2, N);
    k_scatter<OUT_F><<<(int)((E + 7) / 8), 256, 0, stream>>>(srcIdx, dstIdx, dinv, h2, out, E);
    k_post<<<(int)(((long)N * OUT_F + 255) / 256), 256, 0, stream>>>(out, h2, dinv, b2,
                                                                     (long)N * OUT_F);
}